// CS_SS2D_76312978915385
// MI455X (gfx1250) — compile-verified
//
#include <hip/hip_runtime.h>
#include <math.h>

// fp32 WMMA fragment types (16x16x4 f32): A/B = 2 VGPRs/lane, C/D = 8 VGPRs/lane
typedef __attribute__((ext_vector_type(2))) float v2f;
typedef __attribute__((ext_vector_type(8))) float v8f;

// gfx1250 async global->LDS copy pointer types: builtin expects int4 pointers
typedef __attribute__((ext_vector_type(4))) int v4i;
typedef __attribute__((address_space(1))) v4i as1_v4i;
typedef __attribute__((address_space(3))) v4i as3_v4i;

#if defined(__has_builtin)
#  if __has_builtin(__builtin_amdgcn_global_load_async_to_lds_b128)
#    define USE_ASYNC_LDS 1
#  endif
#  if __has_builtin(__builtin_amdgcn_s_wait_asynccnt)
#    define HAVE_WAIT_ASYNC 1
#  endif
#endif

#define B0    8
#define H0    64
#define W0    64
#define L0    4096          // H0*W0
#define DM    96            // d_model
#define DI    192           // d_inner
#define NST   16            // d_state
#define DTR   6             // dt_rank
#define KDIR  4
#define CPROJ 38            // DTR + 2*NST
#define NBL   (B0 * L0)     // 32768 rows

#define NCH   32            // chunks along L for the parallel scan
#define CL    128           // steps per chunk (NCH*CL == L0)

__device__ __forceinline__ float silu_f(float x) { return x / (1.f + __expf(-x)); }
__device__ __forceinline__ float softplus_f(float x) { return (x > 20.f) ? x : log1pf(__expf(x)); }

__device__ __forceinline__ void wait_async_zero()
{
#if USE_ASYNC_LDS
#  if HAVE_WAIT_ASYNC
  __builtin_amdgcn_s_wait_asynccnt(0);
#  else
  asm volatile("s_wait_asynccnt 0x0" ::: "memory");
#  endif
#endif
}

// ---------------------------------------------------------------------------
// GEMM helpers: each wave computes one 16x16 f32 tile via V_WMMA_F32_16X16X4_F32
// A layout (16x4): lanes 0-15 -> M=lane, VGPR0/1 = K0/K1; lanes 16-31 -> K2/K3
// B layout (4x16): VGPR0 lanes0-15 = row K0, lanes16-31 = row K2; VGPR1 = K1/K3
// C/D layout: VGPR r, lanes0-15 -> (M=r, N=lane); lanes16-31 -> (M=r+8, N=lane-16)
// ---------------------------------------------------------------------------

// xz = x[M=32768,96] @ in_proj_w^T[96,384]; split into xm (n<192) and z
__global__ __launch_bounds__(256)
void k_inproj(const float* __restrict__ x, const float* __restrict__ w,
              float* __restrict__ xm, float* __restrict__ zb)
{
  const int lane = threadIdx.x & 31;
  const int wv = blockIdx.x * 8 + (threadIdx.x >> 5);
  const int nt = wv % 24;           // 384/16
  const int mt = wv / 24;           // 32768/16
  const int m0 = mt * 16, n0 = nt * 16;
  const int kh = (lane >> 4) << 1;  // K half: 0 or 2
  const int mr = m0 + (lane & 15);
  const int nc = n0 + (lane & 15);
  v8f acc = {};
  #pragma unroll
  for (int k0 = 0; k0 < DM; k0 += 4) {
    v2f a, b;
    a.x = x[(size_t)mr * DM + k0 + kh];
    a.y = x[(size_t)mr * DM + k0 + kh + 1];
    b.x = w[(size_t)nc * DM + k0 + kh];       // B[k][n] = in_proj_w[n][k]
    b.y = w[(size_t)nc * DM + k0 + kh + 1];
    acc = __builtin_amdgcn_wmma_f32_16x16x4_f32(false, a, false, b, (short)0, acc, false, false);
  }
  const int half = lane >> 4;
  const int n = n0 + (lane & 15);
  #pragma unroll
  for (int r = 0; r < 8; ++r) {
    const int m = m0 + r + half * 8;
    const float v = acc[r];
    if (n < DI) xm[(size_t)m * DI + n] = v;
    else        zb[(size_t)m * DI + (n - DI)] = v;
  }
}

// depthwise 3x3 SAME conv + bias + SiLU; emit hw-order and wh-order copies,
// both with d contiguous so scan-u loads coalesce.
__global__ __launch_bounds__(192)
void k_conv(const float* __restrict__ xm, const float* __restrict__ cw,
            const float* __restrict__ cb,
            float* __restrict__ xcA, float* __restrict__ xcB)
{
  const int m = blockIdx.x;               // b*L + l
  const int c = threadIdx.x;              // channel
  const int b = m >> 12;
  const int l = m & (L0 - 1);
  const int h = l >> 6, w = l & 63;
  float s = cb[c];
  #pragma unroll
  for (int dh = -1; dh <= 1; ++dh) {
    const int hh = h + dh;
    if ((unsigned)hh >= (unsigned)H0) continue;
    #pragma unroll
    for (int dw = -1; dw <= 1; ++dw) {
      const int ww = w + dw;
      if ((unsigned)ww >= (unsigned)W0) continue;
      s += cw[c * 9 + (dh + 1) * 3 + (dw + 1)] *
           xm[(size_t)((b << 12) + (hh << 6) + ww) * DI + c];
    }
  }
  s = silu_f(s);
  xcA[(size_t)((b << 12) + l) * DI + c] = s;
  const int lT = (w << 6) + h;
  xcB[(size_t)((b << 12) + lT) * DI + c] = s;
}

// dbl[b,k,c,l] = sum_d x_proj_weight[k,c,d] * Xdir[l,d]   (c padded 38->48)
__global__ __launch_bounds__(256)
void k_dbl(const float* __restrict__ xcA, const float* __restrict__ xcB,
           const float* __restrict__ xw, float* __restrict__ dbl)
{
  const int lane = threadIdx.x & 31;
  int wv = blockIdx.x * 8 + (threadIdx.x >> 5);   // 24576 waves
  const int lt = wv & 255;  wv >>= 8;             // 4096/16 l-tiles
  const int ct = wv % 3;    wv /= 3;              // 3 c-tiles (48 rows)
  const int k  = wv & 3;
  const int b  = wv >> 2;
  const float* X  = ((k & 1) ? xcB : xcA) + (size_t)b * L0 * DI;
  const float* wk = xw + (size_t)k * CPROJ * DI;
  const int c0 = ct * 16, l0 = lt * 16;
  const int kh = (lane >> 4) << 1;
  const int crow = c0 + (lane & 15);
  const int lcol = l0 + (lane & 15);
  v8f acc = {};
  for (int d0 = 0; d0 < DI; d0 += 4) {
    v2f a = {0.f, 0.f}, bf;
    if (crow < CPROJ) {
      a.x = wk[(size_t)crow * DI + d0 + kh];
      a.y = wk[(size_t)crow * DI + d0 + kh + 1];
    }
    bf.x = X[(size_t)lcol * DI + d0 + kh];
    bf.y = X[(size_t)lcol * DI + d0 + kh + 1];
    acc = __builtin_amdgcn_wmma_f32_16x16x4_f32(false, a, false, bf, (short)0, acc, false, false);
  }
  const int half = lane >> 4;
  const int l = l0 + (lane & 15);
  float* outp = dbl + (size_t)(b * KDIR + k) * CPROJ * L0;
  #pragma unroll
  for (int r = 0; r < 8; ++r) {
    const int c = c0 + r + half * 8;
    if (c < CPROJ) outp[(size_t)c * L0 + l] = acc[r];
  }
}

// Chunked selective scan. h_l = exp(delta_l*A) * h_{l-1} + delta_l*u_l*B_l is a
// first-order linear recurrence -> associative. Pass A (FINAL=false) computes
// per-chunk (prod dA, partial state); pass C (FINAL=true) replays each chunk
// from the prefix-scanned incoming state and emits y via atomic merge of the
// four directions. dt-projection (rank 6) is fused as a wave-uniform dot read
// from LDS-staged rows. The dt/B/C rows are staged with gfx1250 async
// global->LDS b128 copies (ASYNCcnt) when available; LDS layout is row-major
// [c][64] so every copy is a contiguous 256B row segment for all 4 directions
// (reversed directions are staged ascending and indexed with j = 63-t).
// Grid: (bk=32, ch=NCH), 192 threads (one per channel).
template <bool FINAL>
__global__ __launch_bounds__(192)
void k_scan(const float* __restrict__ xcA, const float* __restrict__ xcB,
            const float* __restrict__ dbl,
            const float* __restrict__ dtw,   // [K][DI][DTR]
            const float* __restrict__ dtb,   // [K*DI]
            const float* __restrict__ Alog,  // [K*DI][NST]
            const float* __restrict__ Dsv,   // [K*DI]
            float* __restrict__ pA,          // [bk][ch][n][d] chunk prod(dA)
            float* __restrict__ sB,          // chunk partial-state / h_in
            float* __restrict__ ysum)
{
  const int bk = blockIdx.x;
  const int ch = blockIdx.y;
  const int d  = threadIdx.x;
  const int b  = bk >> 2, k = bk & 3;
  const bool rev = (k >= 2);
  const float* X    = ((k & 1) ? xcB : xcA) + (size_t)b * L0 * DI;
  const float* drow = dbl + (size_t)bk * CPROJ * L0;

  __shared__ float lds[CPROJ * 64];   // 38 rows x 64 steps (256B rows)

  float dtwr[DTR];
  #pragma unroll
  for (int r = 0; r < DTR; ++r) dtwr[r] = dtw[((size_t)k * DI + d) * DTR + r];
  const float bias = dtb[k * DI + d];
  float Ar[NST];
  #pragma unroll
  for (int n = 0; n < NST; ++n) Ar[n] = -__expf(Alog[((size_t)k * DI + d) * NST + n]);
  const float Dk = Dsv[k * DI + d];

  float h[NST], P[NST];
  const size_t sumbase = ((size_t)bk * NCH + ch) * NST * DI + d;
  if (FINAL) {
    #pragma unroll
    for (int n = 0; n < NST; ++n) h[n] = sB[sumbase + (size_t)n * DI];
  } else {
    #pragma unroll
    for (int n = 0; n < NST; ++n) { h[n] = 0.f; P[n] = 1.f; }
  }

  for (int sub = 0; sub < CL / 64; ++sub) {
    const int lbase = ch * CL + sub * 64;
    // staging source base: ascending memory order covering steps of this tile
    const int srcbase = rev ? (L0 - 64 - lbase) : lbase;
    __syncthreads();
    for (int i = threadIdx.x; i < CPROJ * 16; i += 192) {   // 16 float4 per row
      const int c = i >> 4, q = (i & 15) << 2;
      const float* g = drow + (size_t)c * L0 + srcbase + q;
#if USE_ASYNC_LDS
      __builtin_amdgcn_global_load_async_to_lds_b128(
          (as1_v4i*)g, (as3_v4i*)&lds[c * 64 + q], 0, 0);
#else
      const float4 v = *(const float4*)g;
      *(float4*)&lds[c * 64 + q] = v;
#endif
    }
    wait_async_zero();
    __syncthreads();
    for (int t = 0; t < 64; ++t) {
      const int lstep = lbase + t;
      const int jt = rev ? (63 - t) : t;               // index into staged rows
      const int lsrc = srcbase + jt;
      const float u = X[(size_t)lsrc * DI + d];
      float draw = bias;
      #pragma unroll
      for (int r = 0; r < DTR; ++r) draw = fmaf(dtwr[r], lds[r * 64 + jt], draw);
      const float delta = softplus_f(draw);
      const float du = delta * u;
      float y = 0.f;
      #pragma unroll
      for (int n = 0; n < NST; ++n) {
        const float dA = __expf(delta * Ar[n]);
        h[n] = fmaf(h[n], dA, du * lds[(DTR + n) * 64 + jt]);          // * B_l[n]
        if (FINAL) y = fmaf(h[n], lds[(DTR + NST + n) * 64 + jt], y);  // * C_l[n]
        else       P[n] *= dA;
      }
      if (FINAL) {
        y = fmaf(u, Dk, y);
        int ldst;
        if (k == 0)      ldst = lstep;
        else if (k == 2) ldst = L0 - 1 - lstep;
        else {           // wh order -> hw position (+ flip for k==3)
          const int p = (k == 1) ? lstep : (L0 - 1 - lstep);
          ldst = ((p & 63) << 6) + (p >> 6);
        }
        atomicAdd(&ysum[((size_t)b * L0 + ldst) * DI + d], y);
      }
    }
  }
  if (!FINAL) {
    #pragma unroll
    for (int n = 0; n < NST; ++n) {
      pA[sumbase + (size_t)n * DI] = P[n];
      sB[sumbase + (size_t)n * DI] = h[n];
    }
  }
}

// Prefix over chunks per (bk, n, d); rewrites sB in place with each chunk's
// incoming state h_in.
__global__ __launch_bounds__(256)
void k_chscan(const float* __restrict__ pA, float* __restrict__ sB)
{
  const int idx = blockIdx.x * blockDim.x + threadIdx.x;  // 32*16*192 = 98304
  if (idx >= 32 * NST * DI) return;
  const int d  = idx % DI;
  const int n  = (idx / DI) % NST;
  const int bk = idx / (DI * NST);
  float hacc = 0.f;
  for (int ch = 0; ch < NCH; ++ch) {
    const size_t o = (((size_t)bk * NCH + ch) * NST + n) * DI + d;
    const float P = pA[o];
    const float S = sB[o];
    sB[o] = hacc;                 // h_in for chunk ch
    hacc = fmaf(hacc, P, S);
  }
}

// LayerNorm(192) + SiLU(z) gate, one wave per row (wave32 shuffle reduction).
__global__ __launch_bounds__(256)
void k_lngate(float* __restrict__ ysum, const float* __restrict__ zb,
              const float* __restrict__ gamma, const float* __restrict__ beta)
{
  const int lane = threadIdx.x & 31;
  const int row = blockIdx.x * 8 + (threadIdx.x >> 5);
  float v[6]; float s = 0.f, s2 = 0.f;
  #pragma unroll
  for (int i = 0; i < 6; ++i) {
    v[i] = ysum[(size_t)row * DI + i * 32 + lane];
    s += v[i]; s2 += v[i] * v[i];
  }
  #pragma unroll
  for (int off = 16; off; off >>= 1) {
    s  += __shfl_xor(s, off, 32);
    s2 += __shfl_xor(s2, off, 32);
  }
  const float mu  = s * (1.f / DI);
  const float var = s2 * (1.f / DI) - mu * mu;
  const float rstd = rsqrtf(var + 1e-5f);
  #pragma unroll
  for (int i = 0; i < 6; ++i) {
    const int c = i * 32 + lane;
    float yv = (v[i] - mu) * rstd * gamma[c] + beta[c];
    const float zz = zb[(size_t)row * DI + c];
    yv *= silu_f(zz);
    ysum[(size_t)row * DI + c] = yv;
  }
}

// out[m, 0..95] = yg[m, :] @ out_proj_w^T
__global__ __launch_bounds__(256)
void k_outproj(const float* __restrict__ yg, const float* __restrict__ ow,
               float* __restrict__ out)
{
  const int lane = threadIdx.x & 31;
  const int wv = blockIdx.x * 8 + (threadIdx.x >> 5);
  const int nt = wv % 6;            // 96/16
  const int mt = wv / 6;
  const int m0 = mt * 16, n0 = nt * 16;
  const int kh = (lane >> 4) << 1;
  const int mr = m0 + (lane & 15);
  const int nc = n0 + (lane & 15);
  v8f acc = {};
  for (int d0 = 0; d0 < DI; d0 += 4) {
    v2f a, b;
    a.x = yg[(size_t)mr * DI + d0 + kh];
    a.y = yg[(size_t)mr * DI + d0 + kh + 1];
    b.x = ow[(size_t)nc * DI + d0 + kh];      // B[d][c] = out_proj_w[c][d]
    b.y = ow[(size_t)nc * DI + d0 + kh + 1];
    acc = __builtin_amdgcn_wmma_f32_16x16x4_f32(false, a, false, b, (short)0, acc, false, false);
  }
  const int half = lane >> 4;
  const int n = n0 + (lane & 15);
  #pragma unroll
  for (int r = 0; r < 8; ++r) {
    const int m = m0 + r + half * 8;
    out[(size_t)m * DM + n] = acc[r];
  }
}

__global__ void k_zero(float* __restrict__ p, size_t n)
{
  const size_t i = (size_t)blockIdx.x * blockDim.x + threadIdx.x;
  if (i < n) p[i] = 0.f;
}

extern "C" void kernel_launch(void* const* d_in, const int* in_sizes, int n_in,
                              void* d_out, int out_size, void* d_ws, size_t ws_size,
                              hipStream_t stream)
{
  const float* x    = (const float*)d_in[0];
  const float* inw  = (const float*)d_in[1];
  const float* cw   = (const float*)d_in[2];
  const float* cb   = (const float*)d_in[3];
  const float* xw   = (const float*)d_in[4];
  const float* dtw  = (const float*)d_in[5];
  const float* dtb  = (const float*)d_in[6];
  const float* Alog = (const float*)d_in[7];
  const float* Dsv  = (const float*)d_in[8];
  const float* gam  = (const float*)d_in[9];
  const float* bet  = (const float*)d_in[10];
  const float* ow   = (const float*)d_in[11];
  float* out = (float*)d_out;

  // workspace carve-up (~171 MB total; fits in 192 MB L2):
  float* ws  = (float*)d_ws;
  float* xm   = ws;  ws += (size_t)NBL * DI;                   // [b,l,c] pre-conv
  float* zb   = ws;  ws += (size_t)NBL * DI;                   // gate branch
  float* xcA  = ws;  ws += (size_t)NBL * DI;                   // hw order, d contig
  float* xcB  = ws;  ws += (size_t)NBL * DI;                   // wh order, d contig
  float* dbl  = ws;  ws += (size_t)B0 * KDIR * CPROJ * L0;     // dt/B/C rows
  float* ysum = ws;  ws += (size_t)NBL * DI;                   // merged scan output
  float* pA   = ws;  ws += (size_t)32 * NCH * NST * DI;        // chunk prod(dA)
  float* sB   = ws;  ws += (size_t)32 * NCH * NST * DI;        // chunk state / h_in

  const size_t nz = (size_t)NBL * DI;
  k_zero<<<(int)((nz + 255) / 256), 256, 0, stream>>>(ysum, nz);
  k_inproj<<<6144, 256, 0, stream>>>(x, inw, xm, zb);
  k_conv<<<NBL, 192, 0, stream>>>(xm, cw, cb, xcA, xcB);
  k_dbl<<<3072, 256, 0, stream>>>(xcA, xcB, xw, dbl);
  dim3 sg(32, NCH);
  k_scan<false><<<sg, 192, 0, stream>>>(xcA, xcB, dbl, dtw, dtb, Alog, Dsv, pA, sB, ysum);
  k_chscan<<<(32 * NST * DI + 255) / 256, 256, 0, stream>>>(pA, sB);
  k_scan<true><<<sg, 192, 0, stream>>>(xcA, xcB, dbl, dtw, dtb, Alog, Dsv, pA, sB, ysum);
  k_lngate<<<NBL / 8, 256, 0, stream>>>(ysum, zb, gam, bet);
  k_outproj<<<1536, 256, 0, stream>>>(ysum, ow, out);
  (void)in_sizes; (void)n_in; (void)out_size; (void)ws_size;
}